// InfiniteMemory_70205535420770
// MI455X (gfx1250) — compile-verified
//
#include <hip/hip_runtime.h>
#include <math.h>

typedef __attribute__((ext_vector_type(2))) float v2f;
typedef __attribute__((ext_vector_type(8))) float v8f;

#define T_TOK 1024
#define DMODEL 384
#define DK 256
#define DV 256
#define NS 3
#define NH 4
#define CHUNK 16

static __device__ __forceinline__ v8f wmma4(v2f a, v2f b, v8f c) {
  // D = A(16x4 f32) * B(4x16 f32) + C(16x16 f32)
  return __builtin_amdgcn_wmma_f32_16x16x4_f32(false, a, false, b, (short)0, c,
                                               false, false);
}

// out[M=1024][N=256] = emb[1024][384] @ Wp^T   (Wp is [256][384] row-major)
// one wave per 16x16 output tile; 64x16 = 1024 tiles total.
__global__ __launch_bounds__(128) void proj_gemm_kernel(
    const float* __restrict__ emb, const float* __restrict__ Wp,
    float* __restrict__ out) {
  const int wave = (blockIdx.x * blockDim.x + threadIdx.x) >> 5;
  const int tm = wave >> 4;   // 0..63
  const int tn = wave & 15;   // 0..15
  const int lane = threadIdx.x & 31;
  const int half = lane >> 4;
  const int l15 = lane & 15;
  const int m0 = tm * CHUNK, n0 = tn * CHUNK;

  v8f acc = {};
  for (int kk = 0; kk < DMODEL; kk += 4) {
    const int ka = kk + half * 2;
    v2f a, b;
    a.x = emb[(m0 + l15) * DMODEL + ka];
    a.y = emb[(m0 + l15) * DMODEL + ka + 1];
    // B[k][n] = Wp[n][k]
    b.x = Wp[(n0 + l15) * DMODEL + ka];
    b.y = Wp[(n0 + l15) * DMODEL + ka + 1];
    acc = wmma4(a, b, acc);
  }
#pragma unroll
  for (int r = 0; r < 8; ++r)
    out[(m0 + r + half * 8) * DK + n0 + l15] = acc[r];
}

// alpha[t][s], beta[t][s] = sigmoid(k_all[t] . W{a,b}[s,h] + b{a,b}[s,h]).mean(h)
__global__ void gates_kernel(const float* __restrict__ k_all,
                             const float* __restrict__ Wa,
                             const float* __restrict__ ba,
                             const float* __restrict__ Wb,
                             const float* __restrict__ bb,
                             float* __restrict__ alpha,
                             float* __restrict__ beta) {
  const int idx = blockIdx.x * blockDim.x + threadIdx.x;
  if (idx >= T_TOK * NS) return;
  const int t = idx / NS, s = idx % NS;
  const float* kt = k_all + t * DK;
  float sa = 0.f, sb = 0.f;
  for (int h = 0; h < NH; ++h) {
    float da = ba[s * NH + h], db = bb[s * NH + h];
    const float* wa = Wa + (s * NH + h) * DK;
    const float* wb = Wb + (s * NH + h) * DK;
    for (int d = 0; d < DK; ++d) {
      const float kv = kt[d];
      da += kv * wa[d];
      db += kv * wb[d];
    }
    sa += 1.f / (1.f + __expf(-da));
    sb += 1.f / (1.f + __expf(-db));
  }
  alpha[t * NS + s] = sa * 0.25f;
  beta[t * NS + s] = sb * 0.25f;
}

// kn[t] = k_all[t] / max(||k_all[t]||, 1e-12)
__global__ __launch_bounds__(256) void norm_kernel(const float* __restrict__ k_all,
                                                   float* __restrict__ kn) {
  __shared__ float red[256];
  const int t = blockIdx.x, i = threadIdx.x;
  const float v = k_all[t * DK + i];
  red[i] = v * v;
  __syncthreads();
  for (int off = 128; off > 0; off >>= 1) {
    if (i < off) red[i] += red[i + off];
    __syncthreads();
  }
  const float inv = 1.f / fmaxf(sqrtf(red[0]), 1e-12f);
  kn[t * DK + i] = v * inv;
}

// Chunked delta-rule scan. One block per timescale s. 512 threads = 16 waves.
// State W[s] (256x256 f32) lives in global (L2-resident); chunk tiles in LDS.
__global__ __launch_bounds__(512) void scan_kernel(
    const float* __restrict__ kn, const float* __restrict__ v_all,
    const float* __restrict__ alpha, const float* __restrict__ beta,
    float* __restrict__ Wst,    // [NS][DK][DV]
    float* __restrict__ retr) { // [NS][T][DV]
  __shared__ float Kc[CHUNK][DK];  // chunk keys (normalized)
  __shared__ float Vw[CHUNK][DV];  // chunk values, then w rows
  __shared__ float U[CHUNK][DV];   // base = K@W, then retrieved u
  __shared__ float G[CHUNK][CHUNK];
  __shared__ float ga[CHUNK], gb[CHUNK];

  const int s = blockIdx.x;
  float* __restrict__ W = Wst + s * DK * DV;
  const int tid = threadIdx.x;
  const int wv = tid >> 5;  // 0..15
  const int lane = tid & 31;
  const int half = lane >> 4;
  const int l15 = lane & 15;

  // zero the persistent state
  for (int i = tid; i < DK * DV; i += 512) W[i] = 0.f;
  __threadfence();
  __syncthreads();

  for (int c = 0; c < T_TOK / CHUNK; ++c) {
    const int t0 = c * CHUNK;
    // ---- load chunk into LDS ----
    for (int i = tid; i < CHUNK * DK; i += 512) {
      const int r = i >> 8, col = i & 255;
      Kc[r][col] = kn[(t0 + r) * DK + col];
      Vw[r][col] = v_all[(t0 + r) * DV + col];
    }
    if (tid < CHUNK) {
      ga[tid] = alpha[(t0 + tid) * NS + s];
      gb[tid] = beta[(t0 + tid) * NS + s];
    }
    __syncthreads();

    // ---- base = Kc @ W : 16 tiles of 16x16, one per wave ----
    {
      const int n0 = wv * CHUNK;
      v8f acc = {};
      for (int kk = 0; kk < DK; kk += 4) {
        const int ka = kk + half * 2;
        v2f a, b;
        a.x = Kc[l15][ka];
        a.y = Kc[l15][ka + 1];
        b.x = W[ka * DV + n0 + l15];
        b.y = W[(ka + 1) * DV + n0 + l15];
        acc = wmma4(a, b, acc);
      }
#pragma unroll
      for (int r = 0; r < 8; ++r) U[r + half * 8][n0 + l15] = acc[r];
    }
    // ---- Gram G = Kc @ Kc^T (wave 0) ----
    if (wv == 0) {
      v8f acc = {};
      for (int kk = 0; kk < DK; kk += 4) {
        const int ka = kk + half * 2;
        v2f a, b;
        a.x = Kc[l15][ka];
        a.y = Kc[l15][ka + 1];
        b.x = Kc[l15][ka];  // B[k][j] = Kc[j][k], j = l15
        b.y = Kc[l15][ka + 1];
        acc = wmma4(a, b, acc);
      }
#pragma unroll
      for (int r = 0; r < 8; ++r) G[r + half * 8][l15] = acc[r];
    }
    __syncthreads();

    // ---- serial intra-chunk solve (columns independent) ----
    // u_t = base_t + sum_{i<t} G[t,i] * w_i ;  w_t = -a_t*u_t + b_t*v_t
    if (tid < DV) {
      const int col = tid;
      for (int t = 0; t < CHUNK; ++t) {
        const float ut = U[t][col];
        const float wt = -ga[t] * ut + gb[t] * Vw[t][col];
        Vw[t][col] = wt;
        for (int tt = t + 1; tt < CHUNK; ++tt) U[tt][col] += G[tt][t] * wt;
      }
    }
    __syncthreads();

    // ---- retrieved rows out (per-timescale partial) ----
    for (int i = tid; i < CHUNK * DV; i += 512) {
      const int r = i >> 8, col = i & 255;
      retr[(s * T_TOK + t0 + r) * DV + col] = U[r][col];
    }

    // ---- W += Kc^T @ Vw : 256 tiles, 16 per wave ----
    for (int tile = wv; tile < (DK / CHUNK) * (DV / CHUNK); tile += 16) {
      const int m0 = (tile >> 4) * CHUNK, n0 = (tile & 15) * CHUNK;
      v8f acc;
#pragma unroll
      for (int r = 0; r < 8; ++r)
        acc[r] = W[(m0 + r + half * 8) * DV + n0 + l15];
      for (int kk = 0; kk < CHUNK; kk += 4) {
        const int ka = kk + half * 2;
        v2f a, b;
        a.x = Kc[ka][m0 + l15];  // A[m][k] = Kc[k][m]
        a.y = Kc[ka + 1][m0 + l15];
        b.x = Vw[ka][n0 + l15];  // B[k][n] = w_k[n]
        b.y = Vw[ka + 1][n0 + l15];
        acc = wmma4(a, b, acc);
      }
#pragma unroll
      for (int r = 0; r < 8; ++r)
        W[(m0 + r + half * 8) * DV + n0 + l15] = acc[r];
    }
    __threadfence();
    __syncthreads();
  }
}

// out = mean over timescales of retr
__global__ void reduce_kernel(const float* __restrict__ retr,
                              float* __restrict__ out) {
  const int i = blockIdx.x * blockDim.x + threadIdx.x;
  if (i < T_TOK * DV)
    out[i] = (retr[i] + retr[T_TOK * DV + i] + retr[2 * T_TOK * DV + i]) *
             (1.f / 3.f);
}

extern "C" void kernel_launch(void* const* d_in, const int* in_sizes, int n_in,
                              void* d_out, int out_size, void* d_ws,
                              size_t ws_size, hipStream_t stream) {
  const float* emb = (const float*)d_in[0];  // [1024][384]
  const float* Wk  = (const float*)d_in[1];  // [256][384]
  const float* Wv  = (const float*)d_in[2];  // [256][384]
  const float* Wa  = (const float*)d_in[3];  // [3][4][256]
  const float* ba  = (const float*)d_in[4];  // [3][4]
  const float* Wb  = (const float*)d_in[5];  // [3][4][256]
  const float* bb  = (const float*)d_in[6];  // [3][4]
  float* out = (float*)d_out;                // [1024][256]

  // workspace carve-up (~7.1 MB of f32)
  float* k_all = (float*)d_ws;                 // T*DK
  float* v_all = k_all + T_TOK * DK;           // T*DV
  float* kn    = v_all + T_TOK * DV;           // T*DK
  float* alpha = kn + T_TOK * DK;              // T*NS
  float* beta  = alpha + T_TOK * NS;           // T*NS
  float* Wst   = beta + T_TOK * NS;            // NS*DK*DV
  float* retr  = Wst + NS * DK * DV;           // NS*T*DV

  proj_gemm_kernel<<<256, 128, 0, stream>>>(emb, Wk, k_all);
  proj_gemm_kernel<<<256, 128, 0, stream>>>(emb, Wv, v_all);
  gates_kernel<<<(T_TOK * NS + 255) / 256, 256, 0, stream>>>(k_all, Wa, ba, Wb,
                                                             bb, alpha, beta);
  norm_kernel<<<T_TOK, 256, 0, stream>>>(k_all, kn);
  scan_kernel<<<NS, 512, 0, stream>>>(kn, v_all, alpha, beta, Wst, retr);
  reduce_kernel<<<(T_TOK * DV + 255) / 256, 256, 0, stream>>>(retr, out);
}